// TimingMamba_33698313405186
// MI455X (gfx1250) — compile-verified
//
#include <hip/hip_runtime.h>
#include <hip/hip_bf16.h>
#include <math.h>

// ---------------- model constants ----------------
#define D_IN      14
#define D_MODEL   96
#define D_STATE   16
#define D_CONV    4
#define D_INNER   192      // 2 * D_MODEL
#define DT_RANK   6
#define N_LAYERS  2
#define SEQ_LEN   320
#define BATCH     64
#define ROWS      (BATCH * SEQ_LEN)   // 20480 (multiple of 16)
#define DBC_W     (DT_RANK + 2 * D_STATE)   // 38
#define LN_EPS    1e-5f

typedef float v2f __attribute__((ext_vector_type(2)));
typedef float v8f __attribute__((ext_vector_type(8)));

__device__ __forceinline__ float sigmoid_(float x) { return 1.f / (1.f + __expf(-x)); }

// Tail-chunk loader: clamped-address load + select (branch-free; compiles to
// v_cndmask + unconditional load, never reads out of bounds).
template <int K>
__device__ __forceinline__ v2f load2_guard(const float* __restrict__ p, int k, bool ok)
{
    const bool ok0 = ok & (k     < K);
    const bool ok1 = ok & (k + 1 < K);
    const float x0 = p[ok0 ? k     : 0];
    const float x1 = p[ok1 ? k + 1 : 0];
    v2f r;
    r.x = ok0 ? x0 : 0.f;
    r.y = ok1 ? x1 : 0.f;
    return r;
}

// ============================================================
// WMMA fp32 GEMM:  C[M,N] = A[M,K] * W[N,K]^T  (+ fused epilogue)
// One wave per 16x16 tile; K is a compile-time constant so the
// K-loop fully unrolls into a global_load_b64 -> v_wmma stream
// (V_WMMA_F32_16X16X4_F32: full fp32, matches reference exactly).
//   EPI 0: + bias (bias may be null)
//   EPI 1: softplus(acc + bias)                  (dt_proj -> delta)
//   EPI 2: acc + resid  (residual accumulation)  (out_proj -> h)
// ============================================================
template <int K, int EPI>
__global__ __launch_bounds__(256)
void gemm16x16x4(const float* __restrict__ A, int lda,
                 const float* __restrict__ W,        // (N,K) row-major
                 const float* __restrict__ bias,
                 const float* __restrict__ resid,
                 float* __restrict__ C,
                 int M, int N)
{
    const int lane   = threadIdx.x & 31;
    const int wave   = (blockIdx.x * blockDim.x + threadIdx.x) >> 5;
    const int tilesN = (N + 15) >> 4;
    const int tm     = wave / tilesN;
    const int tn     = wave - tm * tilesN;
    if (tm * 16 >= M) return;

    // A operand: lane l -> row (l&15), k-pair base (l>>4)*2  (ISA 16x4 f32 layout)
    // B operand: lane l -> col (l&15), k-pair base (l>>4)*2  (ISA 4x16 layout)
    const int  arow = (tm << 4) + (lane & 15);
    const int  wrow = (tn << 4) + (lane & 15);
    const bool wok  = (wrow < N);
    const int  kofs = (lane >> 4) << 1;

    const float* Ap  = A + (size_t)arow * lda;
    const float* Wp  = W + (size_t)(wok ? wrow : 0) * K;   // clamped: always valid
    const float* Apk = Ap + kofs;                          // 8B-aligned (even strides)
    const float* Wpk = Wp + kofs;

    v8f acc = {};
#pragma unroll
    for (int k0 = 0; k0 < K; k0 += 4) {
        v2f a, b;
        if (k0 + 3 < K) {                       // full chunk: two b64 loads
            a = *(const v2f*)(Apk + k0);
            const v2f bw = *(const v2f*)(Wpk + k0);
            b.x = wok ? bw.x : 0.f;
            b.y = wok ? bw.y : 0.f;
        } else {                                // compile-time tail (K=14, K=6)
            a = load2_guard<K>(Ap, k0 + kofs, true);
            b = load2_guard<K>(Wp, k0 + kofs, wok);
        }
        acc = __builtin_amdgcn_wmma_f32_16x16x4_f32(
                  /*neg_a=*/false, a, /*neg_b=*/false, b,
                  /*c_mod=*/(short)0, acc, /*reuse_a=*/false, /*reuse_b=*/false);
    }

    // C/D layout: VGPR r -> row r (lanes 0-15) / r+8 (lanes 16-31), col = lane&15
    const int ccol = (tn << 4) + (lane & 15);
    if (ccol >= N) return;
    const int   rbase = (tm << 4) + ((lane >> 4) << 3);
    const float bv    = bias ? bias[ccol] : 0.f;
#pragma unroll
    for (int r = 0; r < 8; ++r) {
        const int crow = rbase + r;
        float v = acc[r] + bv;
        if (EPI == 1) v = (v > 20.f) ? v : log1pf(__expf(v));
        if (EPI == 2) v += resid[(size_t)crow * N + ccol];
        C[(size_t)crow * N + ccol] = v;
    }
}

// ============================================================
// LayerNorm over rows of width 96: one wave32 per row.
// ============================================================
__global__ __launch_bounds__(256)
void ln_rows(const float* __restrict__ x, const float* __restrict__ g,
             const float* __restrict__ b, float* __restrict__ out, int rows)
{
    const int wave = (blockIdx.x * blockDim.x + threadIdx.x) >> 5;
    const int lane = threadIdx.x & 31;
    if (wave >= rows) return;
    const float* xr = x + (size_t)wave * D_MODEL;

    float vals[3];
    float s = 0.f;
#pragma unroll
    for (int i = 0; i < 3; ++i) {
        vals[i] = xr[lane + 32 * i];
        s += vals[i];
    }
#pragma unroll
    for (int off = 16; off; off >>= 1) s += __shfl_xor(s, off, 32);
    const float mean = s * (1.f / (float)D_MODEL);

    float v = 0.f;
#pragma unroll
    for (int i = 0; i < 3; ++i) {
        const float d = vals[i] - mean;
        v += d * d;
    }
#pragma unroll
    for (int off = 16; off; off >>= 1) v += __shfl_xor(v, off, 32);
    const float rstd = rsqrtf(v * (1.f / (float)D_MODEL) + LN_EPS);

    float* orow = out + (size_t)wave * D_MODEL;
#pragma unroll
    for (int i = 0; i < 3; ++i) {
        const int idx = lane + 32 * i;
        orow[idx] = (vals[i] - mean) * rstd * g[idx] + b[idx];
    }
}

// ============================================================
// Causal depthwise conv (4 taps) + SiLU.  xh = xz[..., :192].
// ============================================================
__global__ __launch_bounds__(256)
void conv_silu(const float* __restrict__ xz, const float* __restrict__ cw,
               const float* __restrict__ cb, float* __restrict__ xc)
{
    const int idx = blockIdx.x * blockDim.x + threadIdx.x;
    if (idx >= ROWS * D_INNER) return;
    const int c = idx % D_INNER;
    const int t = (idx / D_INNER) % SEQ_LEN;
    const int b = idx / (D_INNER * SEQ_LEN);

    const size_t base = ((size_t)b * SEQ_LEN) * (2 * D_INNER) + c;
    float acc = cb[c];
    if (t >= 3) acc += cw[c * 4 + 0] * xz[base + (size_t)(t - 3) * (2 * D_INNER)];
    if (t >= 2) acc += cw[c * 4 + 1] * xz[base + (size_t)(t - 2) * (2 * D_INNER)];
    if (t >= 1) acc += cw[c * 4 + 2] * xz[base + (size_t)(t - 1) * (2 * D_INNER)];
    acc             += cw[c * 4 + 3] * xz[base + (size_t)t       * (2 * D_INNER)];
    xc[idx] = acc * sigmoid_(acc);
}

// ============================================================
// Selective scan: grid = BATCH, block = D_INNER threads.
// Thread (b,c) keeps h[16] and A[16] in registers; B_t/C_t staged in LDS.
// Output fused with  + u*D  and  * silu(z).
// ============================================================
__global__ __launch_bounds__(D_INNER)
void scan_k(const float* __restrict__ delta, const float* __restrict__ xc,
            const float* __restrict__ dbc,   const float* __restrict__ xz,
            const float* __restrict__ A_log, const float* __restrict__ Dp,
            float* __restrict__ y)
{
    const int b = blockIdx.x;
    const int c = threadIdx.x;

    __shared__ float sB[D_STATE];
    __shared__ float sC[D_STATE];

    float A[D_STATE];
#pragma unroll
    for (int n = 0; n < D_STATE; ++n) A[n] = -__expf(A_log[c * D_STATE + n]);
    const float Dc = Dp[c];

    float h[D_STATE];
#pragma unroll
    for (int n = 0; n < D_STATE; ++n) h[n] = 0.f;

    for (int t = 0; t < SEQ_LEN; ++t) {
        const size_t row = (size_t)b * SEQ_LEN + t;
        if (c < 2 * D_STATE) {
            const float v = dbc[row * DBC_W + DT_RANK + c];
            if (c < D_STATE) sB[c] = v; else sC[c - D_STATE] = v;
        }
        __syncthreads();

        const float d  = delta[row * D_INNER + c];
        const float u  = xc[row * D_INNER + c];
        const float du = d * u;
        float out = 0.f;
#pragma unroll
        for (int n = 0; n < D_STATE; ++n) {
            const float dA = __expf(d * A[n]);
            h[n] = dA * h[n] + du * sB[n];
            out += h[n] * sC[n];
        }
        const float z = xz[row * (2 * D_INNER) + D_INNER + c];
        y[row * D_INNER + c] = (out + u * Dc) * (z * sigmoid_(z));
        __syncthreads();
    }
}

// ============================================================
// Final LN + mean over time: grid = BATCH, block = D_MODEL.
// ============================================================
__global__ __launch_bounds__(D_MODEL)
void pool_ln(const float* __restrict__ h, const float* __restrict__ g,
             const float* __restrict__ b, float* __restrict__ hmean)
{
    const int bb = blockIdx.x;
    const int d  = threadIdx.x;
    __shared__ float red[D_MODEL];
    __shared__ float sm[2];

    float acc = 0.f;
    for (int t = 0; t < SEQ_LEN; ++t) {
        const float v = h[((size_t)bb * SEQ_LEN + t) * D_MODEL + d];
        red[d] = v;
        __syncthreads();
        if (d < 32) {
            float s = red[d] + red[d + 32] + red[d + 64];
#pragma unroll
            for (int off = 16; off; off >>= 1) s += __shfl_xor(s, off, 32);
            if (d == 0) sm[0] = s * (1.f / (float)D_MODEL);
        }
        __syncthreads();
        const float mean = sm[0];
        const float dv   = v - mean;
        red[d] = dv * dv;
        __syncthreads();
        if (d < 32) {
            float s = red[d] + red[d + 32] + red[d + 64];
#pragma unroll
            for (int off = 16; off; off >>= 1) s += __shfl_xor(s, off, 32);
            if (d == 0) sm[1] = s * (1.f / (float)D_MODEL);
        }
        __syncthreads();
        acc += dv * rsqrtf(sm[1] + LN_EPS) * g[d] + b[d];
        __syncthreads();
    }
    hmean[bb * D_MODEL + d] = acc * (1.f / (float)SEQ_LEN);
}

// ============================================================
// Classifier head: 96 -> 32 (ReLU) -> 4.  grid = BATCH, block = 32.
// ============================================================
__global__ __launch_bounds__(32)
void head_k(const float* __restrict__ hmean,
            const float* __restrict__ Wc1, const float* __restrict__ bc1,
            const float* __restrict__ Wc2, const float* __restrict__ bc2,
            float* __restrict__ out)
{
    const int b = blockIdx.x;
    const int j = threadIdx.x;
    __shared__ float h1[32];

    float s = bc1[j];
    for (int d = 0; d < D_MODEL; ++d) s += hmean[b * D_MODEL + d] * Wc1[j * D_MODEL + d];
    h1[j] = fmaxf(s, 0.f);
    __syncthreads();
    if (j < 4) {
        float o = bc2[j];
        for (int k = 0; k < 32; ++k) o += h1[k] * Wc2[j * 32 + k];
        out[b * 4 + j] = o;
    }
}

// ============================================================
// Host orchestration
// ============================================================
static inline int gemm_blocks(int M, int N) {
    const int tiles = (M / 16) * ((N + 15) / 16);
    return (tiles + 7) / 8;                    // 8 waves (256 threads) per block
}

extern "C" void kernel_launch(void* const* d_in, const int* in_sizes, int n_in,
                              void* d_out, int out_size, void* d_ws, size_t ws_size,
                              hipStream_t stream)
{
    const float* x         = (const float*)d_in[0];
    const float* W_in      = (const float*)d_in[1];
    const float* b_in      = (const float*)d_in[2];
    const float* ln_g      = (const float*)d_in[3];
    const float* ln_b      = (const float*)d_in[4];
    const float* in_proj_w = (const float*)d_in[5];
    const float* conv_w    = (const float*)d_in[6];
    const float* conv_b    = (const float*)d_in[7];
    const float* x_proj_w  = (const float*)d_in[8];
    const float* dt_proj_w = (const float*)d_in[9];
    const float* dt_proj_b = (const float*)d_in[10];
    const float* A_log     = (const float*)d_in[11];
    const float* Dp        = (const float*)d_in[12];
    const float* out_projw = (const float*)d_in[13];
    const float* normf_g   = (const float*)d_in[14];
    const float* normf_b   = (const float*)d_in[15];
    const float* Wc1       = (const float*)d_in[16];
    const float* bc1       = (const float*)d_in[17];
    const float* Wc2       = (const float*)d_in[18];
    const float* bc2       = (const float*)d_in[19];
    float* out = (float*)d_out;

    // ---- workspace layout (floats), all L2-resident on MI455X (192MB L2) ----
    float* ws    = (float*)d_ws;
    float* hbuf  = ws;                                  // (ROWS, 96)  residual stream
    float* u     = hbuf  + (size_t)ROWS * D_MODEL;      // (ROWS, 96)  LN output
    float* xz    = u     + (size_t)ROWS * D_MODEL;      // (ROWS, 384)
    float* xc    = xz    + (size_t)ROWS * 2 * D_INNER;  // (ROWS, 192)
    float* dbc   = xc    + (size_t)ROWS * D_INNER;      // (ROWS, 38)
    float* delta = dbc   + (size_t)ROWS * DBC_W;        // (ROWS, 192)
    float* ybuf  = delta + (size_t)ROWS * D_INNER;      // (ROWS, 192)
    float* hmean = ybuf  + (size_t)ROWS * D_INNER;      // (64, 96)

    // 1) embed: h = x @ W_in^T + b_in   (K=14, compile-time tail masking)
    gemm16x16x4<D_IN, 0><<<gemm_blocks(ROWS, D_MODEL), 256, 0, stream>>>(
        x, D_IN, W_in, b_in, nullptr, hbuf, ROWS, D_MODEL);

    for (int l = 0; l < N_LAYERS; ++l) {
        const float* lg   = ln_g      + (size_t)l * D_MODEL;
        const float* lb   = ln_b      + (size_t)l * D_MODEL;
        const float* wip  = in_proj_w + (size_t)l * 2 * D_INNER * D_MODEL;
        const float* cw   = conv_w    + (size_t)l * D_INNER * D_CONV;
        const float* cb   = conv_b    + (size_t)l * D_INNER;
        const float* wxp  = x_proj_w  + (size_t)l * DBC_W * D_INNER;
        const float* wdt  = dt_proj_w + (size_t)l * D_INNER * DT_RANK;
        const float* bdt  = dt_proj_b + (size_t)l * D_INNER;
        const float* Al   = A_log     + (size_t)l * D_INNER * D_STATE;
        const float* Dl   = Dp        + (size_t)l * D_INNER;
        const float* wop  = out_projw + (size_t)l * D_MODEL * D_INNER;

        // 2a) u = LN(h)
        ln_rows<<<(ROWS * 32 + 255) / 256, 256, 0, stream>>>(hbuf, lg, lb, u, ROWS);

        // 2b) xz = u @ in_proj_w^T               (20480 x 384 x 96)
        gemm16x16x4<D_MODEL, 0><<<gemm_blocks(ROWS, 2 * D_INNER), 256, 0, stream>>>(
            u, D_MODEL, wip, nullptr, nullptr, xz, ROWS, 2 * D_INNER);

        // 2c) xc = silu(depthwise_conv(xz[:, :192]))
        conv_silu<<<(ROWS * D_INNER + 255) / 256, 256, 0, stream>>>(xz, cw, cb, xc);

        // 2d) dbc = xc @ x_proj_w^T              (N=38, edge tiles masked)
        gemm16x16x4<D_INNER, 0><<<gemm_blocks(ROWS, DBC_W), 256, 0, stream>>>(
            xc, D_INNER, wxp, nullptr, nullptr, dbc, ROWS, DBC_W);

        // 2e) delta = softplus(dbc[:, :6] @ dt_proj_w^T + dt_proj_b)
        //     (A = dbc with lda=38, K=6 with compile-time tail)
        gemm16x16x4<DT_RANK, 1><<<gemm_blocks(ROWS, D_INNER), 256, 0, stream>>>(
            dbc, DBC_W, wdt, bdt, nullptr, delta, ROWS, D_INNER);

        // 2f) selective scan (fused +u*D and *silu(z))
        scan_k<<<BATCH, D_INNER, 0, stream>>>(delta, xc, dbc, xz, Al, Dl, ybuf);

        // 2g) h = h + y @ out_proj_w^T           (residual fused in epilogue)
        gemm16x16x4<D_INNER, 2><<<gemm_blocks(ROWS, D_MODEL), 256, 0, stream>>>(
            ybuf, D_INNER, wop, nullptr, hbuf, hbuf, ROWS, D_MODEL);
    }

    // 3) final LN + mean over time
    pool_ln<<<BATCH, D_MODEL, 0, stream>>>(hbuf, normf_g, normf_b, hmean);

    // 4) classifier head
    head_k<<<BATCH, 32, 0, stream>>>(hmean, Wc1, bc1, Wc2, bc2, out);
}